// RelGraphConvLayer_3856880632396
// MI455X (gfx1250) — compile-verified
//
#include <hip/hip_runtime.h>
#include <math.h>

#define F 128
#define RREL 3
#define LRELU 0.2f

typedef __attribute__((ext_vector_type(2))) float v2f;
typedef __attribute__((ext_vector_type(8))) float v8f;

// ---------------------------------------------------------------- utilities
__device__ __forceinline__ float waveSum(float v) {
#pragma unroll
    for (int o = 16; o > 0; o >>= 1) v += __shfl_down(v, o, 32);
    return v;
}
__device__ __forceinline__ float waveMax(float v) {
#pragma unroll
    for (int o = 16; o > 0; o >>= 1) v = fmaxf(v, __shfl_down(v, o, 32));
    return v;
}

// ---------------------------------------------------------------- zero output
__global__ void k_zero(float* __restrict__ out, int n) {
    int i = blockIdx.x * blockDim.x + threadIdx.x;
    if (i < n) out[i] = 0.0f;
}

// ------------------------------------------------- per-node attention scores
// s_src[r*N+n] = dot(x[n], att_w[r][:128]);  s_dst[r*N+n] = dot(x[n], att_w[r][128:]) + att_b[r]
__global__ void k_att(const float* __restrict__ x, const float* __restrict__ att_w,
                      const float* __restrict__ att_b, float* __restrict__ s_src,
                      float* __restrict__ s_dst, int N) {
    int lane = threadIdx.x & 31;
    int wave = threadIdx.x >> 5;
    int n = blockIdx.x * 8 + wave;
    int r = blockIdx.y;
    if (n >= N) return;
    const float4* xv = (const float4*)(x + (size_t)n * F);
    const float4* ws = (const float4*)(att_w + (size_t)r * 2 * F);
    const float4* wd = (const float4*)(att_w + (size_t)r * 2 * F + F);
    float4 xl = xv[lane], a = ws[lane], b = wd[lane];
    float ds = xl.x * a.x + xl.y * a.y + xl.z * a.z + xl.w * a.w;
    float dd = xl.x * b.x + xl.y * b.y + xl.z * b.z + xl.w * b.w;
    ds = waveSum(ds);
    dd = waveSum(dd);
    if (lane == 0) {
        s_src[(size_t)r * N + n] = ds;
        s_dst[(size_t)r * N + n] = dd + att_b[r];
    }
}

// ------------------------------------- pass 1: leaky score per edge + block max
__global__ void k_score_max(const float* __restrict__ s_src, const float* __restrict__ s_dst,
                            const int* __restrict__ src, const int* __restrict__ dst,
                            float* __restrict__ aexp, float* __restrict__ partial,
                            int N, int E) {
    __shared__ float sm[8];
    int r = blockIdx.y;
    const int* sr = src + (size_t)r * E;
    const int* dr = dst + (size_t)r * E;
    const float* ss = s_src + (size_t)r * N;
    const float* sd = s_dst + (size_t)r * N;
    float m = -INFINITY;
    int stride = gridDim.x * blockDim.x;
    for (int e = blockIdx.x * blockDim.x + threadIdx.x; e < E; e += stride) {
        float s = ss[sr[e]] + sd[dr[e]];
        s = (s > 0.0f) ? s : LRELU * s;
        aexp[(size_t)r * E + e] = s;
        m = fmaxf(m, s);
    }
    m = waveMax(m);
    int lane = threadIdx.x & 31, wave = threadIdx.x >> 5;
    if (lane == 0) sm[wave] = m;
    __syncthreads();
    if (wave == 0) {
        float v = (lane < 8) ? sm[lane] : -INFINITY;
        v = waveMax(v);
        if (lane == 0) partial[r * 256 + blockIdx.x] = v;
    }
}

__global__ void k_reduce_max(const float* __restrict__ partial, float* __restrict__ rmax) {
    __shared__ float sm[8];
    int r = blockIdx.x;
    float v = partial[r * 256 + threadIdx.x];
    v = waveMax(v);
    int lane = threadIdx.x & 31, wave = threadIdx.x >> 5;
    if (lane == 0) sm[wave] = v;
    __syncthreads();
    if (wave == 0) {
        float w = (lane < 8) ? sm[lane] : -INFINITY;
        w = waveMax(w);
        if (lane == 0) rmax[r] = w;
    }
}

// ------------------------------------- pass 2: exp(score-max), block sums
__global__ void k_score_exp(float* __restrict__ aexp, const float* __restrict__ rmax,
                            float* __restrict__ partial, int E) {
    __shared__ float sm[8];
    int r = blockIdx.y;
    float mx = rmax[r];
    float acc = 0.0f;
    int stride = gridDim.x * blockDim.x;
    for (int e = blockIdx.x * blockDim.x + threadIdx.x; e < E; e += stride) {
        size_t idx = (size_t)r * E + e;
        float v = __expf(aexp[idx] - mx);
        aexp[idx] = v;
        acc += v;
    }
    acc = waveSum(acc);
    int lane = threadIdx.x & 31, wave = threadIdx.x >> 5;
    if (lane == 0) sm[wave] = acc;
    __syncthreads();
    if (wave == 0) {
        float v = (lane < 8) ? sm[lane] : 0.0f;
        v = waveSum(v);
        if (lane == 0) partial[r * 256 + blockIdx.x] = v;
    }
}

__global__ void k_reduce_sum(const float* __restrict__ partial, float* __restrict__ rinv) {
    __shared__ float sm[8];
    int r = blockIdx.x;
    float v = partial[r * 256 + threadIdx.x];
    v = waveSum(v);
    int lane = threadIdx.x & 31, wave = threadIdx.x >> 5;
    if (lane == 0) sm[wave] = v;
    __syncthreads();
    if (wave == 0) {
        float w = (lane < 8) ? sm[lane] : 0.0f;
        w = waveSum(w);
        if (lane == 0) rinv[r] = 1.0f / w;
    }
}

// ------------------------------------------------- node transform GEMM (WMMA)
// xw[n, :] = x[n, :] @ tw_r + tb_r      (one wave computes a 16x128 output tile)
// V_WMMA_F32_16X16X4_F32 fragment layout (ISA 7.12.2):
//   A (16x4):  lane = half*16 + m15 holds A[m15][2*half + j] in vgpr j  -> v2f
//   B (4x16):  lane = half*16 + n15 holds B[2*half + j][n15] in vgpr j  -> v2f
//   C/D (16x16): lane = half*16 + n15, vgpr v holds C[v + 8*half][n15] -> v8f
__global__ void k_gemm_xw(const float* __restrict__ x, const float* __restrict__ tw_r,
                          const float* __restrict__ tb_r, float* __restrict__ xw, int N) {
    int lane = threadIdx.x;          // 32 threads = 1 wave per block
    int m15 = lane & 15;
    int half = lane >> 4;
    int M0 = blockIdx.x * 16;

    int arow = M0 + m15;
    if (arow >= N) arow = N - 1;     // clamp (keeps EXEC all-ones for WMMA)
    const float* arow_p = x + (size_t)arow * F;

    v8f acc[8];
#pragma unroll
    for (int t = 0; t < 8; ++t) acc[t] = (v8f)0.0f;

#pragma unroll 4
    for (int k = 0; k < F; k += 4) {
        // A fragment: two consecutive K values -> one 8-byte load
        v2f a = *(const v2f*)(arow_p + k + 2 * half);
#pragma unroll
        for (int t = 0; t < 8; ++t) {
            v2f b;
            b.x = tw_r[(size_t)(k + 2 * half) * F + t * 16 + m15];
            b.y = tw_r[(size_t)(k + 2 * half + 1) * F + t * 16 + m15];
            acc[t] = __builtin_amdgcn_wmma_f32_16x16x4_f32(
                false, a, false, b, (short)0, acc[t], false, false);
        }
    }

#pragma unroll
    for (int t = 0; t < 8; ++t) {
        float bias = tb_r[t * 16 + m15];
#pragma unroll
        for (int v = 0; v < 8; ++v) {
            int m = M0 + v + 8 * half;
            if (m < N) xw[(size_t)m * F + t * 16 + m15] = acc[t][v] + bias;
        }
    }
}

// ------------------------------------------------- scatter: out[dst] += a*xw[src]
__global__ void k_scatter(const float* __restrict__ xw, const float* __restrict__ aexp_r,
                          const float* __restrict__ rinv_r, const int* __restrict__ src_r,
                          const int* __restrict__ dst_r, float* __restrict__ out, int E) {
    int lane = threadIdx.x & 31;
    int wave = threadIdx.x >> 5;
    int e = blockIdx.x * 8 + wave;
    if (e >= E) return;
    int s = src_r[e];
    int d = dst_r[e];
    float coef = aexp_r[e] * rinv_r[0];
    float4 v = ((const float4*)(xw + (size_t)s * F))[lane];
    float* o = out + (size_t)d * F + lane * 4;
    atomicAdd(o + 0, coef * v.x);
    atomicAdd(o + 1, coef * v.y);
    atomicAdd(o + 2, coef * v.z);
    atomicAdd(o + 3, coef * v.w);
}

// ------------------------------------------------- epilogue: +h_bias, ReLU
__global__ void k_epilogue(float* __restrict__ out, const float* __restrict__ h_bias, int n) {
    int i = blockIdx.x * blockDim.x + threadIdx.x;
    if (i < n) {
        float v = out[i] + h_bias[i & (F - 1)];
        out[i] = v > 0.0f ? v : 0.0f;
    }
}

// ---------------------------------------------------------------- launcher
extern "C" void kernel_launch(void* const* d_in, const int* in_sizes, int n_in,
                              void* d_out, int out_size, void* d_ws, size_t ws_size,
                              hipStream_t stream) {
    const float* x      = (const float*)d_in[0];
    const float* att_w  = (const float*)d_in[1];
    const float* att_b  = (const float*)d_in[2];
    const float* tw     = (const float*)d_in[3];
    const float* tb     = (const float*)d_in[4];
    const float* h_bias = (const float*)d_in[5];
    const int*   src    = (const int*)d_in[6];
    const int*   dst    = (const int*)d_in[7];
    float* out = (float*)d_out;

    const int N = in_sizes[0] / F;
    const int E = in_sizes[6] / RREL;

    // workspace carve-up (floats)
    float* xw      = (float*)d_ws;                    // N*F       (reused per relation)
    float* s_src   = xw + (size_t)N * F;              // R*N
    float* s_dst   = s_src + (size_t)RREL * N;        // R*N
    float* aexp    = s_dst + (size_t)RREL * N;        // R*E
    float* partial = aexp + (size_t)RREL * E;         // R*256
    float* rmax    = partial + RREL * 256;            // R
    float* rinv    = rmax + RREL;                     // R

    const int total = N * F;

    k_zero<<<(total + 255) / 256, 256, 0, stream>>>(out, total);

    k_att<<<dim3((N + 7) / 8, RREL), 256, 0, stream>>>(x, att_w, att_b, s_src, s_dst, N);

    k_score_max<<<dim3(256, RREL), 256, 0, stream>>>(s_src, s_dst, src, dst, aexp, partial, N, E);
    k_reduce_max<<<RREL, 256, 0, stream>>>(partial, rmax);
    k_score_exp<<<dim3(256, RREL), 256, 0, stream>>>(aexp, rmax, partial, E);
    k_reduce_sum<<<RREL, 256, 0, stream>>>(partial, rinv);

    for (int r = 0; r < RREL; ++r) {
        k_gemm_xw<<<(N + 15) / 16, 32, 0, stream>>>(
            x, tw + (size_t)r * F * F, tb + (size_t)r * F, xw, N);
        k_scatter<<<(E + 7) / 8, 256, 0, stream>>>(
            xw, aexp + (size_t)r * E, rinv + r, src + (size_t)r * E, dst + (size_t)r * E, out, E);
    }

    k_epilogue<<<(total + 255) / 256, 256, 0, stream>>>(out, h_bias, total);
}